// kHopAugmentation_23888608100654
// MI455X (gfx1250) — compile-verified
//
#include <hip/hip_runtime.h>

#define NN 4096
#define NNZ (NN * 64)

typedef float v2f __attribute__((ext_vector_type(2)));
typedef float v8f __attribute__((ext_vector_type(8)));

// ---------------- Phase 1: zero workspace (dense A, B) ----------------
__global__ void zero_f4(float4* __restrict__ p, int n4) {
    int i = blockIdx.x * blockDim.x + threadIdx.x;
    if (i < n4) p[i] = make_float4(0.f, 0.f, 0.f, 0.f);
}

// ---------------- Phase 2: COO scatter-add (coalesce duplicates) ------
__global__ void scatter_coo(const int* __restrict__ rows,
                            const int* __restrict__ cols,
                            const float* __restrict__ vals,
                            float* __restrict__ M, int nnz) {
    int i = blockIdx.x * blockDim.x + threadIdx.x;
    if (i < nnz) {
        atomicAdd(M + (size_t)rows[i] * NN + cols[i], vals[i]);
    }
}

// ---------------- Phase 3: f32 WMMA GEMM, double-buffered -------------
// Block: 256 threads = 8 wave32s, computes a 128x128 tile of C.
// Wave grid 2x4: each wave owns 64 rows x 32 cols = 4x2 accumulators of
// v_wmma_f32_16x16x4_f32. K panels of 32 are staged through ping-pong
// LDS buffers (2 x 32KB); next-panel global loads are issued before the
// 64-WMMA compute body and drained into the alternate buffer after it,
// hiding L2/HBM latency behind the matrix pipe.

__device__ __forceinline__ void fetch_panels(const float* __restrict__ A,
                                             const float* __restrict__ B,
                                             int blockRow, int blockCol,
                                             int k0, int tid,
                                             float4 rA[4], float4 rB[4]) {
#pragma unroll
    for (int p = 0; p < 4; ++p) {
        int j = tid + p * 256;
        int rowA = j >> 3, c4A = (j & 7) << 2;       // 128 x 32 panel
        rA[p] = *(const float4*)(A + (size_t)(blockRow + rowA) * NN + k0 + c4A);
        int rowB = j >> 5, c4B = (j & 31) << 2;      // 32 x 128 panel
        rB[p] = *(const float4*)(B + (size_t)(k0 + rowB) * NN + blockCol + c4B);
    }
}

__device__ __forceinline__ void store_panels(float* __restrict__ sA,
                                             float* __restrict__ sB,
                                             int tid,
                                             const float4 rA[4],
                                             const float4 rB[4]) {
#pragma unroll
    for (int p = 0; p < 4; ++p) {
        int j = tid + p * 256;
        int rowA = j >> 3, c4A = (j & 7) << 2;
        *(float4*)(&sA[rowA * 32 + c4A]) = rA[p];
        int rowB = j >> 5, c4B = (j & 31) << 2;
        *(float4*)(&sB[rowB * 128 + c4B]) = rB[p];
    }
}

__global__ __launch_bounds__(256) void gemm_wmma_f32(
        const float* __restrict__ A, const float* __restrict__ B,
        float* __restrict__ C) {
    __shared__ float lA[2][128 * 32];   // ping-pong A panels (2 x 16KB)
    __shared__ float lB[2][32 * 128];   // ping-pong B panels (2 x 16KB)

    const int tid   = threadIdx.x;
    const int lane  = tid & 31;
    const int wave  = tid >> 5;
    const int waveM = wave & 1;      // 0..1 -> 64-row slab
    const int waveN = wave >> 1;     // 0..3 -> 32-col slab
    const int lmod  = lane & 15;     // M (A) / N (B,C) position
    const int lhalf = lane >> 4;     // K-half (A,B) / +8 row (C)

    const int blockRow = blockIdx.y * 128;
    const int blockCol = blockIdx.x * 128;

    v8f acc[4][2];
#pragma unroll
    for (int tm = 0; tm < 4; ++tm)
#pragma unroll
        for (int tn = 0; tn < 2; ++tn) {
            v8f z = {0.f, 0.f, 0.f, 0.f, 0.f, 0.f, 0.f, 0.f};
            acc[tm][tn] = z;
        }

    float4 rA[4], rB[4];
    // Prologue: stage panel 0.
    fetch_panels(A, B, blockRow, blockCol, 0, tid, rA, rB);
    store_panels(lA[0], lB[0], tid, rA, rB);
    __syncthreads();

    for (int k0 = 0; k0 < NN; k0 += 32) {
        const int  buf     = (k0 >> 5) & 1;
        const bool hasNext = (k0 + 32) < NN;

        // Issue next panel's global loads (latency hidden behind WMMAs).
        if (hasNext)
            fetch_panels(A, B, blockRow, blockCol, k0 + 32, tid, rA, rB);
        // Prefetch the panel after next toward the WGP (global_prefetch_b8).
        if (k0 + 64 < NN) {
#pragma unroll
            for (int p = 0; p < 4; ++p) {
                int j = tid + p * 256;
                __builtin_prefetch(
                    A + (size_t)(blockRow + (j >> 3)) * NN + (k0 + 64) + ((j & 7) << 2), 0, 3);
                __builtin_prefetch(
                    B + (size_t)((k0 + 64) + (j >> 5)) * NN + blockCol + ((j & 31) << 2), 0, 3);
            }
        }

        const float* sA = lA[buf];
        const float* sB = lB[buf];
#pragma unroll
        for (int kt = 0; kt < 32; kt += 4) {
            // A fragment (16x4 f32): lanes 0-15 hold M=lmod, K={kt,kt+1};
            // lanes 16-31 hold K={kt+2,kt+3}; contiguous -> ds b64 load.
            v2f afrag[4];
#pragma unroll
            for (int tm = 0; tm < 4; ++tm) {
                const float* ap =
                    &sA[(waveM * 64 + tm * 16 + lmod) * 32 + kt + 2 * lhalf];
                afrag[tm] = *(const v2f*)ap;
            }
            // B fragment (4x16 f32): VGPR0 = row K (kt+2*lhalf), VGPR1 = K+1,
            // N = lmod striped across lanes.
            v2f bfrag[2];
#pragma unroll
            for (int tn = 0; tn < 2; ++tn) {
                int base = (kt + 2 * lhalf) * 128 + waveN * 32 + tn * 16 + lmod;
                v2f b = { sB[base], sB[base + 128] };
                bfrag[tn] = b;
            }
#pragma unroll
            for (int tm = 0; tm < 4; ++tm)
#pragma unroll
                for (int tn = 0; tn < 2; ++tn)
                    acc[tm][tn] = __builtin_amdgcn_wmma_f32_16x16x4_f32(
                        false, afrag[tm], false, bfrag[tn],
                        (short)0, acc[tm][tn], false, false);
        }

        // Drain prefetched regs into the alternate buffer (loads complete
        // here, behind 64 WMMAs of compute).
        if (hasNext)
            store_panels(lA[buf ^ 1], lB[buf ^ 1], tid, rA, rB);
        __syncthreads();
    }

    // Epilogue: C/D layout — VGPR v holds M = v (lanes 0-15) or v+8
    // (lanes 16-31), N = lmod.
#pragma unroll
    for (int tm = 0; tm < 4; ++tm) {
        int rowBase = blockRow + waveM * 64 + tm * 16 + 8 * lhalf;
#pragma unroll
        for (int tn = 0; tn < 2; ++tn) {
            int col = blockCol + waveN * 32 + tn * 16 + lmod;
#pragma unroll
            for (int v = 0; v < 8; ++v)
                C[(size_t)(rowBase + v) * NN + col] = acc[tm][tn][v];
        }
    }
}

extern "C" void kernel_launch(void* const* d_in, const int* in_sizes, int n_in,
                              void* d_out, int out_size, void* d_ws, size_t ws_size,
                              hipStream_t stream) {
    const int*   eiA = (const int*)d_in[0];    // (2, NNZ): rows then cols
    const float* vA  = (const float*)d_in[1];
    const int*   eiB = (const int*)d_in[2];
    const float* vB  = (const float*)d_in[3];

    float* A = (float*)d_ws;                   // 4096*4096 f32 = 64 MB
    float* B = A + (size_t)NN * NN;            // next 64 MB
    float* C = (float*)d_out;

    // Phase 1: zero dense A and B (2 * 16.7M floats = 8.39M float4).
    int n4 = 2 * NN * NN / 4;
    zero_f4<<<n4 / 256, 256, 0, stream>>>((float4*)d_ws, n4);

    // Phase 2: scatter-add edges.
    scatter_coo<<<NNZ / 256, 256, 0, stream>>>(eiA, eiA + NNZ, vA, A, NNZ);
    scatter_coo<<<NNZ / 256, 256, 0, stream>>>(eiB, eiB + NNZ, vB, B, NNZ);

    // Phase 3: WMMA GEMM, 32x32 blocks of 128x128 tiles.
    dim3 grid(NN / 128, NN / 128);
    gemm_wmma_f32<<<grid, 256, 0, stream>>>(A, B, C);
}